// PaiNN_31439160606907
// MI455X (gfx1250) — compile-verified
//
#include <hip/hip_runtime.h>
#include <math.h>

#define NATOMS 10000
#define NEDGES 160000
#define FDIM   128
#define RDIM   20
#define NLAYER 3
#define CUTOFF 5.0f
#define EPS    1e-8f
#define PI_F   3.14159265358979323846f

typedef float v2f __attribute__((ext_vector_type(2)));
typedef float v8f __attribute__((ext_vector_type(8)));

__device__ __forceinline__ float silu_f(float x) {
    return x / (1.0f + __expf(-x));
}

// ---------------------------------------------------------------------------
// WMMA fp32 GEMM:  C[M,Ncols] = act(A[M,K] @ B[K,Ncols] + bias)
//
// Block = (32,4) = 4 waves. Block tile = 64 rows x 64 cols.
// Each wave owns a 16x64 tile (4 v8f accumulators): one 8-byte A-fragment
// load feeds 4 V_WMMA_F32_16X16X4_F32 issues.
//
// B is staged in LDS in *WMMA fragment order*: for k-quad kq and column c,
// the four values B[4kq+0..3][c] are stored contiguously as two lane-half
// pairs at Blds[(kq*64+c)*4 + (r&3)]. A lane's B fragment is then one
// contiguous, 8B-aligned v2f -> a single ds_load_b64, no repack movs.
// Bank index = (4*mlane + 4*half) mod 64: lanes 0-15 hit banks {4m,4m+1},
// lanes 16-31 {4m+4,4m+5} -> banks 4m+2,4m+3 unused, not conflict-free.
// 32 KB static LDS per block (10 blocks/WGP fit in 320 KB).
//
// A fragment per ISA layout: lanes 0-15 -> row M=lane,    K = k0+{0,1}
//                            lanes 16-31 -> row M=lane-16, K = k0+{2,3}
// D layout: VGPR r -> row (r + 8*(lane>=16)), col = lane&15.
// Requires M%16==0, K%128==0, Ncols%64==0 (exact for all calls here).
// ---------------------------------------------------------------------------
#define KCHUNK 128

template <bool BIAS, bool SILU>
__global__ void wmma_gemm_f32(const float* __restrict__ A,
                              const float* __restrict__ B,
                              const float* __restrict__ bias,
                              float* __restrict__ C,
                              int M, int K, int Ncols) {
    __shared__ float Blds[KCHUNK * 64];  // 32 KB, fragment-ordered

    const int lane    = threadIdx.x;                    // 0..31
    const int tid     = threadIdx.y * 32 + lane;        // 0..127
    const int row0    = (blockIdx.x * 4 + threadIdx.y) * 16;
    const int colBase = blockIdx.y * 64;
    const int mlane   = lane & 15;
    const int khalf   = (lane >> 4) * 2;                // 0 / 2 per lane half
    const bool active = (row0 < M);                     // wave-uniform

    const float* Arow = active ? (A + (size_t)(row0 + mlane) * K + khalf)
                               : A;  // dummy, never dereferenced when !active

    v8f acc[4];
#pragma unroll
    for (int t = 0; t < 4; ++t) acc[t] = (v8f){};

    for (int kb = 0; kb < K; kb += KCHUNK) {
        // ---- cooperative stage of B[kb:kb+128, colBase:colBase+64],
        //      scattered into fragment order ----
#pragma unroll 8
        for (int idx = tid; idx < KCHUNK * 64; idx += 128) {
            int r = idx >> 6, c = idx & 63;             // coalesced global read
            Blds[(r >> 2) * 256 + c * 4 + (r & 3)] =
                B[(size_t)(kb + r) * Ncols + colBase + c];
        }
        __syncthreads();

        if (active) {
#pragma unroll 4
            for (int k = 0; k < KCHUNK; k += 4) {
                v2f a = *(const v2f*)(Arow + kb + k);   // K = k+{khalf,khalf+1}
                const float* bq = &Blds[(k >> 2) * 256 + mlane * 4 + khalf * 2];
#pragma unroll
                for (int t = 0; t < 4; ++t) {
                    v2f b = *(const v2f*)(bq + t * 64); // single ds_load_b64
                    acc[t] = __builtin_amdgcn_wmma_f32_16x16x4_f32(
                        false, a, false, b, (short)0, acc[t], false, false);
                }
            }
        }
        __syncthreads();
    }

    if (!active) return;
    const int mBase = row0 + ((lane >> 4) * 8);
#pragma unroll
    for (int t = 0; t < 4; ++t) {
        const int n = colBase + t * 16 + mlane;
        const float bv = BIAS ? bias[n] : 0.0f;
#pragma unroll
        for (int r = 0; r < 8; ++r) {
            float val = acc[t][r] + bv;
            if (SILU) val = silu_f(val);
            C[(size_t)(mBase + r) * Ncols + n] = val;
        }
    }
}

// ---------------------------------------------------------------------------
// Edge geometry: d, r_hat, sinc radial basis, cosine cutoff.
// ---------------------------------------------------------------------------
__global__ void edge_geom_kernel(const float* __restrict__ pos,
                                 const int* __restrict__ esrc,
                                 const int* __restrict__ edst,
                                 float* __restrict__ rbf,
                                 float* __restrict__ fcut,
                                 float* __restrict__ rhat) {
    int e = blockIdx.x * blockDim.x + threadIdx.x;
    if (e >= NEDGES) return;
    int s = esrc[e], d = edst[e];
    float dx = pos[d * 3 + 0] - pos[s * 3 + 0];
    float dy = pos[d * 3 + 1] - pos[s * 3 + 1];
    float dz = pos[d * 3 + 2] - pos[s * 3 + 2];
    float dist = sqrtf(dx * dx + dy * dy + dz * dz + EPS);
    float inv  = 1.0f / dist;
    rhat[e * 3 + 0] = dx * inv;
    rhat[e * 3 + 1] = dy * inv;
    rhat[e * 3 + 2] = dz * inv;
    float arg = PI_F * dist / CUTOFF;
#pragma unroll
    for (int n = 1; n <= RDIM; ++n)
        rbf[(size_t)e * RDIM + (n - 1)] = __sinf((float)n * arg) * inv;
    fcut[e] = 0.5f * (__cosf(arg) + 1.0f) * (dist < CUTOFF ? 1.0f : 0.0f);
}

// ---------------------------------------------------------------------------
// s = embed[atoms], v = 0
// ---------------------------------------------------------------------------
__global__ void init_nodes_kernel(const int* __restrict__ atoms,
                                  const float* __restrict__ embed,
                                  float* __restrict__ s,
                                  float* __restrict__ v) {
    int idx = blockIdx.x * blockDim.x + threadIdx.x;
    if (idx >= NATOMS * FDIM) return;
    int n = idx >> 7, f = idx & (FDIM - 1);
    s[idx] = embed[atoms[n] * FDIM + f];
    size_t vb = (size_t)n * 3 * FDIM + f;
    v[vb] = 0.0f;
    v[vb + FDIM] = 0.0f;
    v[vb + 2 * FDIM] = 0.0f;
}

// ---------------------------------------------------------------------------
// Fused edge message: W = (rbf@rbf_w + b)*fcut recomputed on the fly (never
// materialize [E,384]); vin = pre-scatter snapshot of v (reference gathers
// old v[dst]). One 128-thread block per edge; rbf/r_hat staged in LDS.
// Random-index phi/vin gathers are prefetched (global_prefetch_b8) before
// the 60-MAC rbf dot so the L2 gather overlaps the arithmetic.
// ---------------------------------------------------------------------------
__global__ void msg_edge_kernel(const float* __restrict__ rbf,
                                const float* __restrict__ fcut,
                                const float* __restrict__ rhat,
                                const float* __restrict__ rbf_w,  // [R,3F]
                                const float* __restrict__ rbf_b,  // [3F]
                                const float* __restrict__ phi,    // [N,3F]
                                const float* __restrict__ vin,    // [N,3,F]
                                const int* __restrict__ esrc,
                                const int* __restrict__ edst,
                                float* __restrict__ s,
                                float* __restrict__ v) {
    __shared__ float srbf[RDIM];
    __shared__ float srhat[3];
    const int e = blockIdx.x;
    const int f = threadIdx.x;  // 0..127
    if (f < RDIM) srbf[f] = rbf[(size_t)e * RDIM + f];
    if (f >= 32 && f < 35) srhat[f - 32] = rhat[e * 3 + (f - 32)];
    __syncthreads();

    const int src = esrc[e], dst = edst[e];
    const float fc = fcut[e];

    const size_t pb  = (size_t)dst * 3 * FDIM;
    const size_t vdb = (size_t)dst * 3 * FDIM + f;
    // warm the random-index gather lines while we do the rbf dot
    __builtin_prefetch(&phi[pb + f], 0, 1);
    __builtin_prefetch(&phi[pb + FDIM + f], 0, 1);
    __builtin_prefetch(&phi[pb + 2 * FDIM + f], 0, 1);
    __builtin_prefetch(&vin[vdb], 0, 1);
    __builtin_prefetch(&vin[vdb + FDIM], 0, 1);
    __builtin_prefetch(&vin[vdb + 2 * FDIM], 0, 1);

    float wv = rbf_b[f];
    float wd = rbf_b[FDIM + f];
    float wr = rbf_b[2 * FDIM + f];
#pragma unroll
    for (int r = 0; r < RDIM; ++r) {
        float rb = srbf[r];
        const float* wrow = rbf_w + (size_t)r * 3 * FDIM;
        wv += rb * wrow[f];
        wd += rb * wrow[FDIM + f];
        wr += rb * wrow[2 * FDIM + f];
    }
    wv *= fc; wd *= fc; wr *= fc;

    float gate_v = phi[pb + f] * wv;
    float ds_e   = phi[pb + FDIM + f] * wd;
    float gate_r = phi[pb + 2 * FDIM + f] * wr;

    atomicAdd(&s[(size_t)src * FDIM + f], ds_e);
    const size_t vsb = (size_t)src * 3 * FDIM + f;
#pragma unroll
    for (int i = 0; i < 3; ++i) {
        float dv = vin[vdb + (size_t)i * FDIM] * gate_v + gate_r * srhat[i];
        atomicAdd(&v[vsb + (size_t)i * FDIM], dv);
    }
}

// ---------------------------------------------------------------------------
// Vv_norm + concat([s, Vv_norm]) -> cat [N, 2F]
// ---------------------------------------------------------------------------
__global__ void norm_cat_kernel(const float* __restrict__ s,
                                const float* __restrict__ Vv,
                                float* __restrict__ cat) {
    int idx = blockIdx.x * blockDim.x + threadIdx.x;
    if (idx >= NATOMS * FDIM) return;
    int n = idx >> 7, f = idx & (FDIM - 1);
    size_t vb = (size_t)n * 3 * FDIM + f;
    float a0 = Vv[vb], a1 = Vv[vb + FDIM], a2 = Vv[vb + 2 * FDIM];
    float nrm = sqrtf(a0 * a0 + a1 * a1 + a2 * a2 + EPS);
    cat[(size_t)n * 2 * FDIM + f] = s[idx];
    cat[(size_t)n * 2 * FDIM + FDIM + f] = nrm;
}

// ---------------------------------------------------------------------------
// v += a_vv * Uv ; s += a_sv * <Uv,Vv>_i + a_ss
// ---------------------------------------------------------------------------
__global__ void upd_apply_kernel(const float* __restrict__ a,   // [N,3F]
                                 const float* __restrict__ Uv,  // [N,3,F]
                                 const float* __restrict__ Vv,
                                 float* __restrict__ s,
                                 float* __restrict__ v) {
    int idx = blockIdx.x * blockDim.x + threadIdx.x;
    if (idx >= NATOMS * FDIM) return;
    int n = idx >> 7, f = idx & (FDIM - 1);
    size_t ab = (size_t)n * 3 * FDIM;
    float avv = a[ab + f], asv = a[ab + FDIM + f], ass = a[ab + 2 * FDIM + f];
    size_t vb = ab + f;
    float dot = 0.0f;
#pragma unroll
    for (int i = 0; i < 3; ++i) {
        float u = Uv[vb + (size_t)i * FDIM];
        v[vb + (size_t)i * FDIM] += avv * u;
        dot += u * Vv[vb + (size_t)i * FDIM];
    }
    s[idx] += asv * dot + ass;
}

// ---------------------------------------------------------------------------
// Final GEMV: out[n] = hid[n,:] . w2 + b2
// ---------------------------------------------------------------------------
__global__ void readout_kernel(const float* __restrict__ hid,
                               const float* __restrict__ w2,
                               const float* __restrict__ b2,
                               float* __restrict__ out) {
    int n = blockIdx.x * blockDim.x + threadIdx.x;
    if (n >= NATOMS) return;
    float acc = b2[0];
#pragma unroll 8
    for (int f = 0; f < FDIM; ++f) acc += hid[(size_t)n * FDIM + f] * w2[f];
    out[n] = acc;
}

// ---------------------------------------------------------------------------
extern "C" void kernel_launch(void* const* d_in, const int* in_sizes, int n_in,
                              void* d_out, int out_size, void* d_ws, size_t ws_size,
                              hipStream_t stream) {
    const int*   atoms  = (const int*)  d_in[0];
    const float* pos    = (const float*)d_in[1];
    const int*   esrc   = (const int*)  d_in[2];
    const int*   edst   = (const int*)  d_in[3];
    const float* embed  = (const float*)d_in[4];
    const float* msg_w1 = (const float*)d_in[5];
    const float* msg_b1 = (const float*)d_in[6];
    const float* msg_w2 = (const float*)d_in[7];
    const float* msg_b2 = (const float*)d_in[8];
    const float* rbf_w  = (const float*)d_in[9];
    const float* rbf_b  = (const float*)d_in[10];
    const float* upd_U  = (const float*)d_in[11];
    const float* upd_V  = (const float*)d_in[12];
    const float* upd_w1 = (const float*)d_in[13];
    const float* upd_b1 = (const float*)d_in[14];
    const float* upd_w2 = (const float*)d_in[15];
    const float* upd_b2 = (const float*)d_in[16];
    const float* out_w1 = (const float*)d_in[17];
    const float* out_b1 = (const float*)d_in[18];
    const float* out_w2 = (const float*)d_in[19];
    const float* out_b2 = (const float*)d_in[20];
    float* out = (float*)d_out;

    const size_t NF = (size_t)NATOMS * FDIM;
    float* ws = (float*)d_ws;
    size_t off = 0;
    float* s    = ws + off; off += NF;          // [N,F]
    float* v    = ws + off; off += 3 * NF;      // [N,3,F]
    float* phi  = ws + off; off += 3 * NF;      // [N,3F]  (also 'a' in update)
    float* hid  = ws + off; off += NF;          // [N,F]
    float* Uv   = ws + off; off += 3 * NF;      // [N,3,F] (also vin snapshot)
    float* Vv   = ws + off; off += 3 * NF;      // [N,3,F]
    float* cat  = ws + off; off += 2 * NF;      // [N,2F]
    float* rbfb = ws + off; off += (size_t)NEDGES * RDIM;
    float* fcut = ws + off; off += (size_t)NEDGES;
    float* rhat = ws + off; off += (size_t)NEDGES * 3;

    const dim3 wblk(32, 4);                 // 4 waves: 64-row block tile
    const int gx_N  = (NATOMS + 63) / 64;   // 157 (last block partially idle)
    const int gx_3N = (3 * NATOMS + 63) / 64;  // 469
    const int gy_F  = FDIM / 64;            // 2
    const int gy_3F = 3 * FDIM / 64;        // 6

    edge_geom_kernel<<<(NEDGES + 255) / 256, 256, 0, stream>>>(
        pos, esrc, edst, rbfb, fcut, rhat);
    init_nodes_kernel<<<(NATOMS * FDIM + 255) / 256, 256, 0, stream>>>(
        atoms, embed, s, v);

    for (int l = 0; l < NLAYER; ++l) {
        const float* mw1 = msg_w1 + (size_t)l * FDIM * FDIM;
        const float* mb1 = msg_b1 + (size_t)l * FDIM;
        const float* mw2 = msg_w2 + (size_t)l * FDIM * 3 * FDIM;
        const float* mb2 = msg_b2 + (size_t)l * 3 * FDIM;
        const float* rw  = rbf_w  + (size_t)l * RDIM * 3 * FDIM;
        const float* rb  = rbf_b  + (size_t)l * 3 * FDIM;
        const float* uU  = upd_U  + (size_t)l * FDIM * FDIM;
        const float* uV  = upd_V  + (size_t)l * FDIM * FDIM;
        const float* uw1 = upd_w1 + (size_t)l * 2 * FDIM * FDIM;
        const float* ub1 = upd_b1 + (size_t)l * FDIM;
        const float* uw2 = upd_w2 + (size_t)l * FDIM * 3 * FDIM;
        const float* ub2 = upd_b2 + (size_t)l * 3 * FDIM;

        // ---- message block ----
        wmma_gemm_f32<true, true><<<dim3(gx_N, gy_F), wblk, 0, stream>>>(
            s, mw1, mb1, hid, NATOMS, FDIM, FDIM);
        wmma_gemm_f32<true, false><<<dim3(gx_N, gy_3F), wblk, 0, stream>>>(
            hid, mw2, mb2, phi, NATOMS, FDIM, 3 * FDIM);
        // snapshot v (reference gathers pre-update v[dst])
        hipMemcpyAsync(Uv, v, sizeof(float) * 3 * NF,
                       hipMemcpyDeviceToDevice, stream);
        msg_edge_kernel<<<NEDGES, FDIM, 0, stream>>>(
            rbfb, fcut, rhat, rw, rb, phi, Uv, esrc, edst, s, v);

        // ---- update block ----
        wmma_gemm_f32<false, false><<<dim3(gx_3N, gy_F), wblk, 0, stream>>>(
            v, uU, nullptr, Uv, 3 * NATOMS, FDIM, FDIM);
        wmma_gemm_f32<false, false><<<dim3(gx_3N, gy_F), wblk, 0, stream>>>(
            v, uV, nullptr, Vv, 3 * NATOMS, FDIM, FDIM);
        norm_cat_kernel<<<(NATOMS * FDIM + 255) / 256, 256, 0, stream>>>(
            s, Vv, cat);
        wmma_gemm_f32<true, true><<<dim3(gx_N, gy_F), wblk, 0, stream>>>(
            cat, uw1, ub1, hid, NATOMS, 2 * FDIM, FDIM);
        wmma_gemm_f32<true, false><<<dim3(gx_N, gy_3F), wblk, 0, stream>>>(
            hid, uw2, ub2, phi, NATOMS, FDIM, 3 * FDIM);
        upd_apply_kernel<<<(NATOMS * FDIM + 255) / 256, 256, 0, stream>>>(
            phi, Uv, Vv, s, v);
    }

    // ---- readout ----
    wmma_gemm_f32<true, true><<<dim3(gx_N, gy_F), wblk, 0, stream>>>(
        s, out_w1, out_b1, hid, NATOMS, FDIM, FDIM);
    readout_kernel<<<(NATOMS + 255) / 256, 256, 0, stream>>>(
        hid, out_w2, out_b2, out);
}